// HashGridTcnn_79164837200475
// MI455X (gfx1250) — compile-verified
//
#include <hip/hip_runtime.h>
#include <hip/hip_bf16.h>
#include <cmath>

#define L_LEVELS 16
#define T_SIZE   65536u
#define COLS     71          // 32 hash feats + 39 positional-encoding feats
#define WAVES    8
#define BLOCK    (WAVES * 32)
#define TILE_F   (32 * COLS) // 2272 floats = 9088 bytes per wave tile

struct ResTable { float r[L_LEVELS]; };

__global__ __launch_bounds__(BLOCK) void hashgrid_pe_kernel(
    const float* __restrict__ x, const float* __restrict__ tin,
    const int* __restrict__ mask, const int* __restrict__ layeridp,
    const float* __restrict__ tables, const float* __restrict__ bbox,
    float* __restrict__ out, int N, ResTable res)
{
    extern __shared__ float smem[];                 // WAVES * TILE_F floats
    const int lane      = threadIdx.x & 31;
    const int wave      = threadIdx.x >> 5;
    const int wave_base = blockIdx.x * BLOCK + wave * 32;
    if (wave_base >= N) return;
    const bool full_wave = (wave_base + 32) <= N;   // always true for N%32==0
    const int n  = wave_base + lane;
    const int nc = (n < N) ? n : (N - 1);

    float* wt  = smem + (size_t)wave * TILE_F;      // this wave's tile
    float* row = wt + lane * COLS;                  // stride 71 (odd) -> no bank conflicts

    // ---- x_t = concat(x[:, keep], t) ----
    float xt[3] = {0.f, 0.f, 0.f};
    {
        const float xv[3] = { x[3*(size_t)nc + 0], x[3*(size_t)nc + 1], x[3*(size_t)nc + 2] };
        int k = 0;
        #pragma unroll
        for (int d = 0; d < 3; ++d)
            if (mask[d] != 0 && k < 3) { xt[k] = xv[d]; ++k; }
        if (k < 3) { xt[k] = tin[nc]; ++k; }
    }

    // ---- normalize to [0,1] ----
    float x01[3];
    #pragma unroll
    for (int d = 0; d < 3; ++d) {
        const float lo = bbox[d], hi = bbox[3 + d];
        const float v = (xt[d] - lo) / (hi - lo);
        x01[d] = fminf(fmaxf(v, 0.f), 1.f);
    }

    // ---- multires hash grid: 16 levels x 8 corners, tables L2-resident (8MB) ----
    const int layerid = layeridp[0];
    const float* tab0 = tables + (size_t)layerid * ((size_t)L_LEVELS * T_SIZE * 2u);
    #pragma unroll
    for (int l = 0; l < L_LEVELS; ++l) {
        const float r = res.r[l];
        const float px = x01[0]*r, py = x01[1]*r, pz = x01[2]*r;
        const float fx = floorf(px), fy = floorf(py), fz = floorf(pz);
        const float dx = px - fx, dy = py - fy, dz = pz - fz;
        const unsigned ix = (unsigned)fx, iy = (unsigned)fy, iz = (unsigned)fz;
        const unsigned hx0 = ix;                  const unsigned hx1 = ix + 1u;
        const unsigned hy0 = iy * 2654435761u;    const unsigned hy1 = hy0 + 2654435761u;
        const unsigned hz0 = iz * 805459861u;     const unsigned hz1 = hz0 + 805459861u;
        const float* tl = tab0 + (size_t)l * (T_SIZE * 2u);
        float acc0 = 0.f, acc1 = 0.f;
        #pragma unroll
        for (int c = 0; c < 8; ++c) {             // offs[c] = (c>>2, c>>1&1, c&1)
            const unsigned h = ((c & 4) ? hx1 : hx0) ^ ((c & 2) ? hy1 : hy0)
                             ^ ((c & 1) ? hz1 : hz0);
            const unsigned idx = h & (T_SIZE - 1u);
            const float2 f2 = *(const float2*)(tl + (size_t)idx * 2u); // global_load_b64
            const float w = ((c & 4) ? dx : 1.f - dx)
                          * ((c & 2) ? dy : 1.f - dy)
                          * ((c & 1) ? dz : 1.f - dz);
            acc0 = fmaf(w, f2.x, acc0);
            acc1 = fmaf(w, f2.y, acc1);
        }
        row[2*l + 0] = acc0;
        row[2*l + 1] = acc1;
    }

    // ---- positional encoding: [x_t, (sin, cos)(x_t * 2^i * pi), i=0..5] ----
    const float PI_F = 3.14159274101257324f;      // float(pi), matches f32 promotion
    #pragma unroll
    for (int d = 0; d < 3; ++d) row[32 + d] = xt[d];
    #pragma unroll
    for (int i = 0; i < 6; ++i) {
        #pragma unroll
        for (int d = 0; d < 3; ++d) {
            const float arg = (xt[d] * (float)(1 << i)) * PI_F;
            float s, c;
            __sincosf(arg, &s, &c);
            row[35 + 6*i + d] = s;
            row[38 + 6*i + d] = c;
        }
    }

    if (full_wave) {
        // Fence LDS writes, then stream the 9088B tile to HBM with the CDNA5
        // async store-from-LDS path (ASYNCcnt); S_ENDPGM's implicit wait-idle drains it.
        asm volatile("s_wait_dscnt 0x0" ::: "memory");
        const unsigned long long gbase =
            (unsigned long long)(uintptr_t)out +
            (unsigned long long)__builtin_amdgcn_readfirstlane(wave_base) * (COLS * 4ull);
        const unsigned lds_base = (unsigned)(uintptr_t)wt;
        // 17 x B128 rounds: lane j covers bytes [(j*32+lane)*16, +16)  (16B aligned)
        #pragma unroll 1
        for (int j = 0; j < 17; ++j) {
            const unsigned off = (unsigned)(j * 32 + lane) * 16u;
            asm volatile("global_store_async_from_lds_b128 %0, %1, %2"
                         :: "v"(off), "v"(lds_base + off), "s"(gbase) : "memory");
        }
        // remaining 96 floats: 3 x B32 rounds
        #pragma unroll 1
        for (int j = 0; j < 3; ++j) {
            const unsigned off = 8704u + (unsigned)(j * 32 + lane) * 4u;
            asm volatile("global_store_async_from_lds_b32 %0, %1, %2"
                         :: "v"(off), "v"(lds_base + off), "s"(gbase) : "memory");
        }
    } else if (n < N) {
        #pragma unroll 1
        for (int k = 0; k < COLS; ++k)
            out[(size_t)n * COLS + k] = row[k];
    }
}

extern "C" void kernel_launch(void* const* d_in, const int* in_sizes, int n_in,
                              void* d_out, int out_size, void* d_ws, size_t ws_size,
                              hipStream_t stream) {
    (void)n_in; (void)d_ws; (void)ws_size; (void)out_size;
    const float* x       = (const float*)d_in[0];
    const float* t       = (const float*)d_in[1];
    const int*   mask    = (const int*)  d_in[2];
    const int*   layerid = (const int*)  d_in[3];
    const float* tables  = (const float*)d_in[4];
    const float* bbox    = (const float*)d_in[5];
    float*       out     = (float*)d_out;
    const int N = in_sizes[1];                    // t is [N,1]

    // Replicate numpy's double-precision res table exactly (floor sits near integers).
    ResTable res;
    const double b = std::exp((std::log(512.0) - std::log(16.0)) / 15.0);
    for (int l = 0; l < L_LEVELS; ++l)
        res.r[l] = (float)std::floor(16.0 * std::pow(b, (double)l));

    const int blocks = (N + BLOCK - 1) / BLOCK;
    const size_t shmem = (size_t)WAVES * TILE_F * sizeof(float); // 72,704 B
    hipLaunchKernelGGL(hashgrid_pe_kernel, dim3(blocks), dim3(BLOCK), shmem, stream,
                       x, t, mask, layerid, tables, bbox, out, N, res);
}